// GRU_43301860278767
// MI455X (gfx1250) — compile-verified
//
#include <hip/hip_runtime.h>

// ---------------------------------------------------------------------------
// GRU on gfx1250 (MI455X), fp32 WMMA (V_WMMA_F32_16X16X4_F32).
//
// - Recurrence is independent per batch row: 64 rows -> 4 independent 16-row
//   groups -> 4 workgroups, zero device-wide sync (only s_barrier in-block).
// - Recurrent state (h, z, r*h) lives in LDS; only hidden_out stores leave.
// - Weights pre-transposed once per launch -> contiguous K for B fragments.
// - K permuted in groups of 8 so ONE b128 load feeds TWO WMMAs per operand
//   (A and B use the same permutation; sum over K unchanged).
// ---------------------------------------------------------------------------

typedef float v2f __attribute__((ext_vector_type(2)));
typedef float v4f __attribute__((ext_vector_type(4)));
typedef float v8f __attribute__((ext_vector_type(8)));

#define T_SEQ  512
#define BATCH  64
#define DIN    256
#define DLAT   512
#define DOUTD  256
#define KTOT   (DIN + DLAT)            // 768
#define TPB    512                     // 16 waves per block (wave32)
#define LDS_STRIDE 516                 // 512+4: 16B-aligned rows, distinct banks

__device__ __forceinline__ float fast_sigmoid(float x) {
    return __builtin_amdgcn_rcpf(1.0f + __expf(-x));
}
__device__ __forceinline__ float fast_tanh(float x) {
    // tanh(x) = 1 - 2/(exp(2x)+1): exp->inf gives 1, exp->0 gives -1 (stable)
    float e = __expf(2.0f * x);
    return 1.0f - 2.0f * __builtin_amdgcn_rcpf(e + 1.0f);
}

// Two 16x16x4 fp32 WMMAs fed by one v4f per operand (8 K-values, permuted).
__device__ __forceinline__ v8f wmma2(v8f c, v4f a, v4f b) {
    v2f aA = { a.x, a.y }, aB = { a.z, a.w };
    v2f bA = { b.x, b.y }, bB = { b.z, b.w };
    c = __builtin_amdgcn_wmma_f32_16x16x4_f32(false, aA, false, bA,
                                              (short)0, c, false, false);
    c = __builtin_amdgcn_wmma_f32_16x16x4_f32(false, aB, false, bB,
                                              (short)0, c, false, false);
    return c;
}

// out[n*K + k] = in[k*N + n]   (row-major [K,N] -> [N,K])
__global__ __launch_bounds__(256)
void transpose_kernel(const float* __restrict__ in, float* __restrict__ out,
                      int K, int N) {
    int idx = blockIdx.x * 256 + threadIdx.x;      // over K*N
    if (idx < K * N) {
        int n = idx / K;
        int k = idx - n * K;
        out[idx] = in[(size_t)k * N + n];          // writes coalesced
    }
}

// ---------------------------------------------------------------------------
// Sequential recurrence: one block per 16 batch rows, 16 waves per block,
// wave w owns N-tiles 2w and 2w+1 of the [16, 512] gate outputs.
// ---------------------------------------------------------------------------
__global__ __launch_bounds__(TPB)
void gru_seq_kernel(const float* __restrict__ x,    // [T, B, DIN]
                    const float* __restrict__ WzT,  // [DLAT][KTOT] transposed
                    const float* __restrict__ WrT,
                    const float* __restrict__ WnT,
                    const float* __restrict__ bz,
                    const float* __restrict__ br,
                    const float* __restrict__ bn,
                    float* __restrict__ hidden_out) // [T, B, DLAT]
{
    __shared__ float hS [16 * LDS_STRIDE];   // current hidden state
    __shared__ float zS [16 * LDS_STRIDE];   // z gate
    __shared__ float rhS[16 * LDS_STRIDE];   // r * h

    const unsigned tid  = threadIdx.x;
    const unsigned lane = tid & 31u;
    const unsigned half = lane >> 4;               // lane group (K sub-pairs)
    const unsigned lr   = lane & 15u;
    const unsigned wave = tid >> 5;                // 0..15
    const unsigned mrow = blockIdx.x * 16u + lr;   // global batch row (A side)
    const unsigned koff = 4u * half;               // b128 offset within K-group

    for (unsigned i = tid; i < 16u * LDS_STRIDE; i += TPB) {
        hS[i] = 0.0f; zS[i] = 0.0f; rhS[i] = 0.0f;
    }
    __syncthreads();

    const unsigned col0 = (2u * wave    ) * 16u + lr;
    const unsigned col1 = (2u * wave + 1) * 16u + lr;
    const float bzc[2] = { bz[col0], bz[col1] };
    const float brc[2] = { br[col0], br[col1] };
    const float bnc[2] = { bn[col0], bn[col1] };

#pragma unroll 1
    for (int t = 0; t < T_SEQ; ++t) {
        const float* xt = x + (size_t)t * BATCH * DIN + (size_t)mrow * DIN;

        // ------------- z and r gates for both owned N-tiles -------------
#pragma unroll
        for (int s = 0; s < 2; ++s) {
            const unsigned bcol = (2u * wave + (unsigned)s) * 16u + lr;
            const float* wzp = WzT + (size_t)bcol * KTOT + koff;
            const float* wrp = WrT + (size_t)bcol * KTOT + koff;
            v8f cz = {};
            v8f cr = {};
            // x part of K (global, L2-hot)
#pragma unroll 4
            for (int k0 = 0; k0 < DIN; k0 += 8) {
                v4f a  = *(const v4f*)(xt + k0 + koff);
                v4f wz = *(const v4f*)(wzp + k0);
                v4f wr = *(const v4f*)(wrp + k0);
                cz = wmma2(cz, a, wz);
                cr = wmma2(cr, a, wr);
            }
            // h part of K (LDS)
#pragma unroll 4
            for (int k0 = DIN; k0 < KTOT; k0 += 8) {
                v4f a  = *(const v4f*)(&hS[lr * LDS_STRIDE + (k0 - DIN) + koff]);
                v4f wz = *(const v4f*)(wzp + k0);
                v4f wr = *(const v4f*)(wrp + k0);
                cz = wmma2(cz, a, wz);
                cr = wmma2(cr, a, wr);
            }
#pragma unroll
            for (int v = 0; v < 8; ++v) {
                const unsigned row = (unsigned)v + 8u * half;
                const unsigned idx = row * LDS_STRIDE + bcol;
                const float zv = fast_sigmoid(cz[v] + bzc[s]);
                const float rv = fast_sigmoid(cr[v] + brc[s]);
                zS[idx]  = zv;
                rhS[idx] = rv * hS[idx];
            }
        }

        __syncthreads();   // zS / rhS complete within the block

        // ------------- n gate + h update for both owned N-tiles -------------
        float* hout = hidden_out + (size_t)t * BATCH * DLAT;
#pragma unroll
        for (int s = 0; s < 2; ++s) {
            const unsigned bcol = (2u * wave + (unsigned)s) * 16u + lr;
            const float* wnp = WnT + (size_t)bcol * KTOT + koff;
            v8f cn = {};
#pragma unroll 4
            for (int k0 = 0; k0 < DIN; k0 += 8) {
                v4f a  = *(const v4f*)(xt + k0 + koff);
                v4f wn = *(const v4f*)(wnp + k0);
                cn = wmma2(cn, a, wn);
            }
#pragma unroll 4
            for (int k0 = DIN; k0 < KTOT; k0 += 8) {
                v4f a  = *(const v4f*)(&rhS[lr * LDS_STRIDE + (k0 - DIN) + koff]);
                v4f wn = *(const v4f*)(wnp + k0);
                cn = wmma2(cn, a, wn);
            }
#pragma unroll
            for (int v = 0; v < 8; ++v) {
                const unsigned row = (unsigned)v + 8u * half;
                const unsigned idx = row * LDS_STRIDE + bcol;
                const float ht = fast_tanh(cn[v] + bnc[s]);
                const float zv = zS[idx];
                const float hv = hS[idx];
                const float hn = (1.0f - zv) * hv + zv * ht;
                hS[idx] = hn;   // in place: this wave owns these (row, col) cells
                hout[(size_t)(blockIdx.x * 16u + row) * DLAT + bcol] = hn;
            }
        }

        __syncthreads();   // hS update complete before next step's reads
    }
}

// ---------------------------------------------------------------------------
// Output projection: outputs = hidden @ Wo + bo.  One wave per 16x16 tile.
// ---------------------------------------------------------------------------
__global__ __launch_bounds__(256)
void gru_out_kernel(const float* __restrict__ hidden,   // [T*B, DLAT]
                    const float* __restrict__ WoT,      // [DOUTD][DLAT] transposed
                    const float* __restrict__ bo,       // [DOUTD]
                    float* __restrict__ out)            // [T*B, DOUTD]
{
    const unsigned tid   = threadIdx.x;
    const unsigned lane  = tid & 31u;
    const unsigned half  = lane >> 4;
    const unsigned lr    = lane & 15u;
    const unsigned wave  = (blockIdx.x * 256u + tid) >> 5;
    const unsigned mtile = wave >> 4;           // 0..2047
    const unsigned ntile = wave & 15u;          // 0..15
    const unsigned arow  = mtile * 16u + lr;
    const unsigned bcol  = ntile * 16u + lr;
    const unsigned koff  = 4u * half;

    const float* ap  = hidden + (size_t)arow * DLAT + koff;
    const float* wop = WoT + (size_t)bcol * DLAT + koff;
    v8f c = {};
#pragma unroll 4
    for (int k0 = 0; k0 < DLAT; k0 += 8) {
        v4f a = *(const v4f*)(ap + k0);
        v4f b = *(const v4f*)(wop + k0);
        c = wmma2(c, a, b);
    }
    const float bb = bo[bcol];
#pragma unroll
    for (int v = 0; v < 8; ++v) {
        const unsigned mrow = mtile * 16u + (unsigned)v + 8u * half;
        out[(size_t)mrow * DOUTD + bcol] = c[v] + bb;
    }
}

// ---------------------------------------------------------------------------
extern "C" void kernel_launch(void* const* d_in, const int* in_sizes, int n_in,
                              void* d_out, int out_size, void* d_ws, size_t ws_size,
                              hipStream_t stream) {
    const float* x  = (const float*)d_in[0];
    const float* Wz = (const float*)d_in[1];
    const float* bz = (const float*)d_in[2];
    const float* Wr = (const float*)d_in[3];
    const float* br = (const float*)d_in[4];
    const float* Wn = (const float*)d_in[5];
    const float* bn = (const float*)d_in[6];
    const float* Wo = (const float*)d_in[7];
    const float* bo = (const float*)d_in[8];

    float* outputs = (float*)d_out;                                   // [T,B,DOUTD]
    float* hidden  = outputs + (size_t)T_SEQ * BATCH * DOUTD;         // [T,B,DLAT]

    float* ws  = (float*)d_ws;
    float* WzT = ws;                                   // [DLAT][KTOT]  1.5 MB
    float* WrT = WzT + (size_t)DLAT * KTOT;
    float* WnT = WrT + (size_t)DLAT * KTOT;
    float* WoT = WnT + (size_t)DLAT * KTOT;            // [DOUTD][DLAT] 0.5 MB

    const int nzr = KTOT * DLAT;                       // 393216
    const int no  = DLAT * DOUTD;                      // 131072
    transpose_kernel<<<(nzr + 255) / 256, 256, 0, stream>>>(Wz, WzT, KTOT, DLAT);
    transpose_kernel<<<(nzr + 255) / 256, 256, 0, stream>>>(Wr, WrT, KTOT, DLAT);
    transpose_kernel<<<(nzr + 255) / 256, 256, 0, stream>>>(Wn, WnT, KTOT, DLAT);
    transpose_kernel<<<(no  + 255) / 256, 256, 0, stream>>>(Wo, WoT, DLAT, DOUTD);

    // 4 independent batch groups, no device-wide sync needed.
    gru_seq_kernel<<<4, TPB, 0, stream>>>(x, WzT, WrT, WnT, bz, br, bn, hidden);

    // 32768 tiles, 8 waves per block -> 4096 blocks.
    gru_out_kernel<<<4096, 256, 0, stream>>>(hidden, WoT, bo, outputs);
}